// QuantIntSoft_61959198212654
// MI455X (gfx1250) — compile-verified
//
#include <hip/hip_runtime.h>
#include <math.h>

// I-BERT int-softmax constants (from reference)
#define COEF0 0.35815147f
#define COEF1 0.96963238f
#define COEF2 1.0f
#define X0    (-0.6931f)
#define NSHIFT 30
#define ROW_LEN 2048
#define TPB 256
#define EPT 8   // ROW_LEN / TPB

typedef float v4f __attribute__((ext_vector_type(4)));

// ---- CDNA5 async global->LDS copy (16B per lane), non-temporal --------------
// LDS byte address in a VGPR; global address = SGPR base + VGPR byte offset
__device__ __forceinline__ void async_copy16_nt(unsigned lds_byte_off,
                                                const void* gbase_uniform,
                                                unsigned gbyte_off) {
    asm volatile("global_load_async_to_lds_b128 %0, %1, %2 offset:0 th:TH_LOAD_NT"
                 :
                 : "v"(lds_byte_off), "v"(gbyte_off), "s"(gbase_uniform)
                 : "memory");
}

__device__ __forceinline__ void wait_async0() {
#if __has_builtin(__builtin_amdgcn_s_wait_asynccnt)
    __builtin_amdgcn_s_wait_asynccnt(0);
#else
    asm volatile("s_wait_asynccnt 0" ::: "memory");
#endif
}

// ---- wave32 butterfly reductions via ds_swizzle (XOR pattern in immediate) --
// offset = xor<<10 | or<<5 | and : SWAPX16=0x401f SWAPX8=0x201f SWAPX4=0x101f
//                                  SWAPX2=0x081f SWAPX1=0x041f
template <int PAT>
__device__ __forceinline__ float swz(float v) {
    return __int_as_float(__builtin_amdgcn_ds_swizzle(__float_as_int(v), PAT));
}
__device__ __forceinline__ float wave_max32(float v) {
    v = fmaxf(v, swz<0x401f>(v));
    v = fmaxf(v, swz<0x201f>(v));
    v = fmaxf(v, swz<0x101f>(v));
    v = fmaxf(v, swz<0x081f>(v));
    v = fmaxf(v, swz<0x041f>(v));
    return v;
}
__device__ __forceinline__ float wave_sum32(float v) {
    v += swz<0x401f>(v);
    v += swz<0x201f>(v);
    v += swz<0x101f>(v);
    v += swz<0x081f>(v);
    v += swz<0x041f>(v);
    return v;
}

__global__ __launch_bounds__(TPB)
void ibert_softmax_row_kernel(const float* __restrict__ x,
                              const float* __restrict__ scale,
                              float* __restrict__ out) {
    __shared__ __align__(16) float row[ROW_LEN];
    __shared__ __align__(16) float red[16];   // [0..7] wave maxes, [8..15] wave sums

    const int tid  = threadIdx.x;
    const int lane = tid & 31;             // wave32
    const int wave = tid >> 5;
    const long long rbase = (long long)blockIdx.x * ROW_LEN;
    const float* __restrict__ gin = x + rbase;

    // ---- Stage one 8KB row into LDS via the async engine --------------------
    const unsigned lds_base = (unsigned)(uintptr_t)(&row[0]);
    const unsigned o0 = (unsigned)tid * 16u;
    const unsigned o1 = o0 + (unsigned)(TPB * 16);   // +4096B
    async_copy16_nt(lds_base + o0, gin, o0);
    async_copy16_nt(lds_base + o1, gin, o1);
    wait_async0();
    __syncthreads();

    // ---- Uniform scalar prep (hoisted; once per 2048 elements) --------------
    const float s      = scale[0];
    const float inv_s  = 1.0f / s;
    const float b_int  = floorf((COEF1 / COEF0) * inv_s);
    const float c_int  = floorf((COEF2 / COEF0) * inv_s * inv_s);
    const float x0_int = floorf(X0 * inv_s);         // negative integer-valued
    const float inv_x0 = 1.0f / x0_int;
    const float clampv = (float)NSHIFT * x0_int;

    // ---- Phase 1: row max over RAW x (monotone: max(x*c)==max(x)*c, c>0) ----
    float xr[EPT];
    {
        const v4f* rowv = (const v4f*)row;
        v4f a = rowv[2 * tid];
        v4f b = rowv[2 * tid + 1];
        xr[0] = a.x; xr[1] = a.y; xr[2] = a.z; xr[3] = a.w;
        xr[4] = b.x; xr[5] = b.y; xr[6] = b.z; xr[7] = b.w;
    }
    float m = fmaxf(fmaxf(fmaxf(xr[0], xr[1]), fmaxf(xr[2], xr[3])),
                    fmaxf(fmaxf(xr[4], xr[5]), fmaxf(xr[6], xr[7])));
    m = wave_max32(m);
    if (lane == 0) red[wave] = m;
    __syncthreads();
    {
        const v4f* redv = (const v4f*)red;
        v4f r0 = redv[0], r1 = redv[1];
        m = fmaxf(fmaxf(fmaxf(r0.x, r0.y), fmaxf(r0.z, r0.w)),
                  fmaxf(fmaxf(r1.x, r1.y), fmaxf(r1.z, r1.w)));
    }
    const float neg_m = -(m * inv_s);                // -(row max of x/s)

    // ---- Phase 2: integer exp polynomial + row sum --------------------------
    float ei[EPT];
    float sum = 0.0f;
#pragma unroll
    for (int k = 0; k < EPT; ++k) {
        float v = fmaxf(fmaf(xr[k], inv_s, neg_m), clampv); // x/s - max, clamped
        float q = floorf(v * inv_x0);                // halvings in [0, N]
        float r = fmaf(-x0_int, q, v);               // remainder in (x0_int, 0]
        float z = fmaf(r, r + b_int, c_int);         // int polynomial
        float e = floorf(ldexpf(z, NSHIFT - (int)q)); // z * 2^(N-q)
        e = fmaxf(e, 0.0f);
        ei[k] = e;
        sum += e;
    }
    sum = wave_sum32(sum);
    if (lane == 0) red[8 + wave] = sum;
    __syncthreads();
    {
        const v4f* redv = (const v4f*)red;
        v4f r0 = redv[2], r1 = redv[3];
        sum = ((r0.x + r0.y) + (r0.z + r0.w)) + ((r1.x + r1.y) + (r1.z + r1.w));
    }

    // ---- Phase 3: inverse softmax -> nearest power of two (bit domain) ------
    float o[EPT];
#pragma unroll
    for (int k = 0; k < EPT; ++k) {
        float rec = __builtin_amdgcn_rcpf(ei[k]);    // ei==0 -> +inf
        float v   = rintf(sum * rec);                // round(sum/ei), v >= 1
        unsigned b = __float_as_uint(v);
        // rounds = floor(log2 v) + (mant >= 1.5): exponent bits + mantissa MSB
        int rounds = (int)(b >> 23) - 127 + (int)((b >> 22) & 1u);
        int qlog = rounds < 0 ? 0 : (rounds > 255 ? 255 : rounds);
        float oo = ldexpf(1.0f, -qlog);              // 2^-qlog (handles denorms)
        // mask: inf/nan (ei==0). rounds>=256 is unreachable for finite f32.
        if (b >= 0x7f800000u) oo = 0.0f;
        o[k] = oo;
    }
    float* __restrict__ gout = out + rbase + (long long)tid * EPT;
    v4f oa = {o[0], o[1], o[2], o[3]};
    v4f ob = {o[4], o[5], o[6], o[7]};
    __builtin_nontemporal_store(oa, (v4f*)gout);      // streaming store
    __builtin_nontemporal_store(ob, (v4f*)(gout + 4));
}

extern "C" void kernel_launch(void* const* d_in, const int* in_sizes, int n_in,
                              void* d_out, int out_size, void* d_ws, size_t ws_size,
                              hipStream_t stream) {
    const float* x     = (const float*)d_in[0];
    const float* scale = (const float*)d_in[1];
    float* out         = (float*)d_out;

    const int rows = in_sizes[0] / ROW_LEN;          // 2*12*2048 = 49152
    dim3 grid((unsigned)rows), block(TPB);
    ibert_softmax_row_kernel<<<grid, block, 0, stream>>>(x, scale, out);
}